// DifferentiableProcessor_8315056685436
// MI455X (gfx1250) — compile-verified
//
#include <hip/hip_runtime.h>
#include <math.h>

#define H_IMG 1024
#define W_IMG 1536
#define B_IMG 2
#define HW ((size_t)H_IMG * (size_t)W_IMG)

typedef __attribute__((ext_vector_type(2))) float v2f;
typedef __attribute__((ext_vector_type(8))) float v8f;

__device__ __forceinline__ float clip01(float v) { return fminf(1.f, fmaxf(0.f, v)); }
__device__ __forceinline__ float pmod1(float v) { return v - floorf(v); }
__device__ __forceinline__ float pmod(float v, float d) { float m = fmodf(v, d); return m < 0.f ? m + d : m; }

// normalized 1D gaussian tap (matches _gauss2d's separable factor)
__device__ float gaussw(int k, int size, float sigma) {
    int c = size / 2;
    float inv = 1.f / (2.f * sigma * sigma);
    float sum = 0.f;
    for (int j = 0; j < size; ++j) { float d = (float)(j - c); sum += expf(-d * d * inv); }
    float d = (float)(k - c);
    return expf(-d * d * inv) / sum;
}

// ---------------------------------------------------------------------------
// K1: fused pointwise chain: exposure -> gamma2.2 -> contrast -> gamma ->
//     HSL band edits -> saturation/vibrance -> dehaze. Writes X (3ch) + luma.
// ---------------------------------------------------------------------------
__global__ void k1_pointwise(const float* __restrict__ x,
                             const float* exposure, const float* contrast, const float* gamma,
                             const float* hue_shifts, const float* sat_mults, const float* lum_shifts,
                             const float* saturation, const float* vibrance, const float* dehaze,
                             float* __restrict__ X, float* __restrict__ luma_out) {
    size_t idx = (size_t)blockIdx.x * blockDim.x + threadIdx.x;
    size_t npx = (size_t)B_IMG * HW;
    if (idx >= npx) return;
    size_t b = idx / HW, px = idx % HW;
    const float* base = x + (b * 3) * HW + px;
    float r = base[0], g = base[HW], bl = base[2 * HW];

    float ex = exp2f(fminf(4.f, fmaxf(-3.f, exposure[0])));
    r *= ex; g *= ex; bl *= ex;
    const float ig = 1.f / 2.2f;
    r = powf(fmaxf(r, 1e-6f), ig); g = powf(fmaxf(g, 1e-6f), ig); bl = powf(fmaxf(bl, 1e-6f), ig);
    float cf = 1.f + tanhf(contrast[0]) * 0.3f;
    r = 0.5f + (r - 0.5f) * cf; g = 0.5f + (g - 0.5f) * cf; bl = 0.5f + (bl - 0.5f) * cf;
    float ge = 1.f + tanhf(gamma[0]) * 0.2f;
    r = powf(fminf(1.f, fmaxf(r, 1e-6f)), ge);
    g = powf(fminf(1.f, fmaxf(g, 1e-6f)), ge);
    bl = powf(fminf(1.f, fmaxf(bl, 1e-6f)), ge);
    r = clip01(r); g = clip01(g); bl = clip01(bl);

    // rgb -> hsl
    float mx = fmaxf(r, fmaxf(g, bl)), mn = fminf(r, fminf(g, bl));
    float delta = mx - mn;
    float l = 0.5f * (mx + mn);
    float s = (delta < 1e-6f) ? 0.f : delta / (1.f - fabsf(2.f * l - 1.f) + 1e-6f);
    float h = 0.f;
    if (delta > 1e-6f) {           // last-where-wins priority: b > g > r
        if (mx == bl)      h = (r - g) / (delta + 1e-6f) + 4.f;
        else if (mx == g)  h = (bl - r) / (delta + 1e-6f) + 2.f;
        else               h = pmod((g - bl) / (delta + 1e-6f), 6.f);
    }
    h = pmod1(h / 6.f);

    // 8 hue-band edits
    const float centers[8] = {0.f, 0.083f, 0.167f, 0.333f, 0.5f, 0.667f, 0.75f, 0.917f};
    float ths = 0.f, tsm = 1.f, tls = 0.f;
    const float winv = 1.f / (2.f * 0.08f * 0.08f);
#pragma unroll
    for (int i = 0; i < 8; ++i) {
        float hd = h - centers[i];
        float dist = fminf(fabsf(hd), fminf(fabsf(hd - 1.f), fabsf(hd + 1.f)));
        float w = expf(-dist * dist * winv) * s;
        ths += hue_shifts[i] * 0.1f * w;
        tsm += (sat_mults[i] - 1.f) * w;
        tls += lum_shifts[i] * 0.2f * w;
    }
    float h2 = pmod1(h + ths);
    float s2 = clip01(s * tsm);
    float l2 = clip01(l + tls);

    // hsl -> rgb
    float c = (1.f - fabsf(2.f * l2 - 1.f)) * s2;
    float hp = h2 * 6.f;
    float xx = c * (1.f - fabsf(fmodf(hp, 2.f) - 1.f));
    float r1 = 0.f, g1 = 0.f, b1 = 0.f;
    int hi = (int)hp; if (hi > 5) hi = 5;
    switch (hi) {
        case 0: r1 = c;  g1 = xx; break;
        case 1: r1 = xx; g1 = c;  break;
        case 2: g1 = c;  b1 = xx; break;
        case 3: g1 = xx; b1 = c;  break;
        case 4: r1 = xx; b1 = c;  break;
        default: r1 = c; b1 = xx; break;
    }
    float m = l2 - 0.5f * c;
    r = clip01(r1 + m); g = clip01(g1 + m); bl = clip01(b1 + m);

    // saturation / vibrance
    float lum = 0.2126f * r + 0.7152f * g + 0.0722f * bl;
    float mx2 = fmaxf(r, fmaxf(g, bl)), mn2 = fminf(r, fminf(g, bl));
    float cur = (mx2 - mn2) / (mx2 + 1e-6f);
    float total = (1.f + tanhf(saturation[0]) * 0.5f) * (1.f + tanhf(vibrance[0]) * 0.5f * (1.f - cur));
    r = clip01(lum + total * (r - lum));
    g = clip01(lum + total * (g - lum));
    bl = clip01(lum + total * (bl - lum));

    // dehaze
    float amt = tanhf(dehaze[0]) * 0.5f;
    float dark = fminf(r, fminf(g, bl));
    if (amt > 0.f) {
        float inv = 1.f / (1.f - amt + 1e-6f);
        r = clip01((r - dark * amt) * inv);
        g = clip01((g - dark * amt) * inv);
        bl = clip01((bl - dark * amt) * inv);
    } else {
        r = clip01(r * (1.f + amt) - amt * 0.5f);
        g = clip01(g * (1.f + amt) - amt * 0.5f);
        bl = clip01(bl * (1.f + amt) - amt * 0.5f);
    }

    float* xo = X + (b * 3) * HW + px;
    xo[0] = r; xo[HW] = g; xo[2 * HW] = bl;
    luma_out[idx] = 0.2126f * r + 0.7152f * g + 0.0722f * bl;
}

// ---------------------------------------------------------------------------
// Horizontal separable conv: 256-px row segment staged in LDS with halo.
// Halo staging uses clamped loads + value select (no exec-masked loads).
// ---------------------------------------------------------------------------
__global__ void hconv_dual_kernel(const float* __restrict__ in,
                                  float* __restrict__ out31, float* __restrict__ out7) {
    const int PAD = 15, SEG = 256;
    __shared__ float tile[SEG + 2 * PAD];
    __shared__ float w31[31], w7[7];
    int t = threadIdx.x;
    if (t < 31) w31[t] = gaussw(t, 31, 8.f);
    if (t < 7)  w7[t]  = gaussw(t, 7, 1.5f);
    int segs = W_IMG / SEG;
    int bid = blockIdx.x;
    int seg = bid % segs; int rest = bid / segs;
    int row = rest % H_IMG; int p = rest / H_IMG;
    const float* rp = in + (size_t)p * HW + (size_t)row * W_IMG;
    int basec = seg * SEG - PAD;
    for (int i = t; i < SEG + 2 * PAD; i += 256) {
        int xc = basec + i;
        int xcc = min(max(xc, 0), W_IMG - 1);
        float v = rp[xcc];
        tile[i] = (xc == xcc) ? v : 0.f;
    }
    __syncthreads();
    float a31 = 0.f, a7 = 0.f;
#pragma unroll
    for (int k = 0; k < 31; ++k) a31 += w31[k] * tile[t + k];
#pragma unroll
    for (int k = 0; k < 7; ++k)  a7  += w7[k]  * tile[t + 12 + k];
    size_t o = (size_t)p * HW + (size_t)row * W_IMG + seg * SEG + t;
    out31[o] = a31; out7[o] = a7;
}

template <int TAPS>
__global__ void hconv_kernel(const float* __restrict__ in, float* __restrict__ out, float sigma) {
    const int PAD = TAPS / 2, SEG = 256;
    __shared__ float tile[SEG + 2 * PAD];
    __shared__ float w[TAPS];
    int t = threadIdx.x;
    if (t < TAPS) w[t] = gaussw(t, TAPS, sigma);
    int segs = W_IMG / SEG;
    int bid = blockIdx.x;
    int seg = bid % segs; int rest = bid / segs;
    int row = rest % H_IMG; int p = rest / H_IMG;
    const float* rp = in + (size_t)p * HW + (size_t)row * W_IMG;
    int basec = seg * SEG - PAD;
    for (int i = t; i < SEG + 2 * PAD; i += 256) {
        int xc = basec + i;
        int xcc = min(max(xc, 0), W_IMG - 1);
        float v = rp[xcc];
        tile[i] = (xc == xcc) ? v : 0.f;
    }
    __syncthreads();
    float a = 0.f;
#pragma unroll
    for (int k = 0; k < TAPS; ++k) a += w[k] * tile[t + k];
    out[(size_t)p * HW + (size_t)row * W_IMG + seg * SEG + t] = a;
}

// ---------------------------------------------------------------------------
// Vertical separable conv on a 16x16 tile via exact-f32 WMMA 16x16x4:
//   O[i][j] = sum_e band(i,e) * I[r0-pad+e][c0+j], band chunked into K=4 slices.
// A layout (16x4 f32): lanes0-15 VGPR{0,1}=K{0,1}; lanes16-31 VGPR{0,1}=K{2,3}.
// B layout (4x16 f32): symmetric (lane half selects K pair, lane&15 = N).
//
// A-weights come from a zero-padded LDS table wpad[16+d]=w[d] (always
// in-bounds, no compares). Interior tiles (uniform scalar test) take a fast
// path: one base pointer + constant row-stride instruction offsets, no clamps.
// Border tiles use clamped loads + value select.
// ---------------------------------------------------------------------------
template <int TAPS>
__device__ __forceinline__ v8f vconv_wmma(const float* __restrict__ plane,
                                          int r0, int c0,
                                          const float* __restrict__ wpad, v8f acc) {
    constexpr int PAD = TAPS / 2;
    constexpr int NCH = (TAPS + 15 + 3) >> 2;
    int lane = threadIdx.x & 31;
    int m = lane & 15;
    int koff = (lane >> 4) << 1;
    // wl[4*chk + j] = wpad[16 + (4*chk + koff + j) - m] = band(m, 4*chk+koff+j)
    const float* wl = wpad + 16 + koff - m;
    // uniform (scalar) interior test: entire band row range in-bounds
    bool interior = (r0 >= PAD) && (r0 - PAD + 4 * NCH - 1 < H_IMG);
    if (interior) {
        const float* cptr = plane + (size_t)(r0 - PAD + koff) * W_IMG + c0 + m;
#pragma unroll
        for (int chk = 0; chk < NCH; ++chk) {
            float a0 = wl[4 * chk];
            float a1 = wl[4 * chk + 1];
            float b0 = cptr[(size_t)(4 * chk) * W_IMG];
            float b1 = cptr[(size_t)(4 * chk + 1) * W_IMG];
            v2f av; av.x = a0; av.y = a1;
            v2f bv; bv.x = b0; bv.y = b1;
            acc = __builtin_amdgcn_wmma_f32_16x16x4_f32(false, av, false, bv, (short)0, acc,
                                                        false, false);
        }
    } else {
        const float* cptr = plane + c0 + m;
#pragma unroll
        for (int chk = 0; chk < NCH; ++chk) {
            float a0 = wl[4 * chk];
            float a1 = wl[4 * chk + 1];
            int row0 = r0 - PAD + 4 * chk + koff;
            int row1 = row0 + 1;
            int rc0 = min(max(row0, 0), H_IMG - 1);
            int rc1 = min(max(row1, 0), H_IMG - 1);
            float b0 = cptr[(size_t)rc0 * W_IMG];
            float b1 = cptr[(size_t)rc1 * W_IMG];
            b0 = (rc0 == row0) ? b0 : 0.f;
            b1 = (rc1 == row1) ? b1 : 0.f;
            v2f av; av.x = a0; av.y = a1;
            v2f bv; bv.x = b0; bv.y = b1;
            acc = __builtin_amdgcn_wmma_f32_16x16x4_f32(false, av, false, bv, (short)0, acc,
                                                        false, false);
        }
    }
    return acc;
}

// fill a zero-padded banded weight table: wpad[16+d] = w[d], 0 elsewhere
template <int TAPS, int TABN>
__device__ __forceinline__ void fill_wpad(float* wpad, float sigma) {
    int t = threadIdx.x;
    for (int i = t; i < TABN; i += blockDim.x) {
        int d = i - 16;
        wpad[i] = (d >= 0 && d < TAPS) ? gaussw(d, TAPS, sigma) : 0.f;
    }
}

// decompose a wave-uniform tile id through readfirstlane so tile math + the
// interior branch stay scalar (SALU + s_cbranch; EXEC untouched around WMMA)
__device__ __forceinline__ void tile_coords(int& p, int& r0, int& c0) {
    const int tilesX = W_IMG / 16, tilesY = H_IMG / 16;
    int wave = blockIdx.x * (blockDim.x >> 5) + (threadIdx.x >> 5);
    wave = __builtin_amdgcn_readfirstlane(wave);
    int tx = wave % tilesX; int rest = wave / tilesX;
    int ty = rest % tilesY; p = rest / tilesY;
    r0 = ty * 16; c0 = tx * 16;
}

// K3: vertical g31+g7 on luma + clarity/texture ratio; updates X, luma in place.
__global__ void k3_clarity(const float* __restrict__ h31, const float* __restrict__ h7,
                           float* __restrict__ luma, float* __restrict__ X,
                           const float* clarity, const float* texture) {
    __shared__ float wp31[16 + 48];   // 16 zeros + 4*NCH(12) slots
    __shared__ float wp7[16 + 24];    // 16 zeros + 4*NCH(6) slots
    fill_wpad<31, 64>(wp31, 8.f);
    fill_wpad<7, 40>(wp7, 1.5f);
    __syncthreads();
    int p, r0, c0;
    tile_coords(p, r0, c0);
    v8f acc31 = {0.f, 0.f, 0.f, 0.f, 0.f, 0.f, 0.f, 0.f};
    acc31 = vconv_wmma<31>(h31 + (size_t)p * HW, r0, c0, wp31, acc31);
    v8f acc7 = {0.f, 0.f, 0.f, 0.f, 0.f, 0.f, 0.f, 0.f};
    acc7 = vconv_wmma<7>(h7 + (size_t)p * HW, r0, c0, wp7, acc7);
    float ck = tanhf(clarity[0]) * 0.5f;
    float tk = tanhf(texture[0]) * 0.3f;
    int lane = threadIdx.x & 31, m = lane & 15;
    int rbase = r0 + ((lane >> 4) << 3);
#pragma unroll
    for (int v = 0; v < 8; ++v) {
        int row = rbase + v, col = c0 + m;
        size_t off = (size_t)p * HW + (size_t)row * W_IMG + col;
        float lu = luma[off];
        float le = lu + ck * (lu - acc31[v]) + tk * (lu - acc7[v]);
        float ratio = (le + 1e-6f) / (lu + 1e-6f);
        float* xp = X + (size_t)(p * 3) * HW + (size_t)row * W_IMG + col;
        float r = clip01(xp[0] * ratio);
        float g = clip01(xp[HW] * ratio);
        float b = clip01(xp[2 * HW] * ratio);
        xp[0] = r; xp[HW] = g; xp[2 * HW] = b;
        luma[off] = 0.2126f * r + 0.7152f * g + 0.0722f * b;   // luma2
    }
}

// K5: vertical g7 on luma2 + unsharp mask; writes x3 to out.
__global__ void k5_sharpen(const float* __restrict__ h7, const float* __restrict__ luma2,
                           const float* __restrict__ X, float* __restrict__ out,
                           const float* sharpen) {
    __shared__ float wp7[16 + 24];
    fill_wpad<7, 40>(wp7, 1.5f);
    __syncthreads();
    int p, r0, c0;
    tile_coords(p, r0, c0);
    v8f acc = {0.f, 0.f, 0.f, 0.f, 0.f, 0.f, 0.f, 0.f};
    acc = vconv_wmma<7>(h7 + (size_t)p * HW, r0, c0, wp7, acc);
    float s_amt = 1.f / (1.f + expf(-sharpen[0]));
    bool passthru = s_amt < 0.01f;
    int lane = threadIdx.x & 31, m = lane & 15;
    int rbase = r0 + ((lane >> 4) << 3);
#pragma unroll
    for (int v = 0; v < 8; ++v) {
        int row = rbase + v, col = c0 + m;
        size_t off = (size_t)p * HW + (size_t)row * W_IMG + col;
        float lu = luma2[off];
        float sharp = lu + s_amt * (lu - acc[v]);
        float ratio = fminf(2.f, fmaxf(0.5f, (sharp + 1e-6f) / (lu + 1e-6f)));
        size_t xoff = (size_t)(p * 3) * HW + (size_t)row * W_IMG + col;
#pragma unroll
        for (int ch = 0; ch < 3; ++ch) {
            float xv = X[xoff + ch * HW];
            float xs = clip01(xv * ratio);
            out[xoff + ch * HW] = passthru ? xv : xs;
        }
    }
}

// K7: vertical g51 per plane + Orton screen blend, in place on out.
__global__ void k7_orton(const float* __restrict__ h51, float* __restrict__ out,
                         const float* orton) {
    __shared__ float wp51[16 + 68];   // 16 zeros + 4*NCH(17) slots
    fill_wpad<51, 84>(wp51, 15.f);
    __syncthreads();
    int p, r0, c0;                     // p in [0, 6)
    tile_coords(p, r0, c0);
    v8f acc = {0.f, 0.f, 0.f, 0.f, 0.f, 0.f, 0.f, 0.f};
    acc = vconv_wmma<51>(h51 + (size_t)p * HW, r0, c0, wp51, acc);
    float o_amt = 0.4f / (1.f + expf(-orton[0]));
    bool passthru = o_amt < 0.01f;
    int lane = threadIdx.x & 31, m = lane & 15;
    int rbase = r0 + ((lane >> 4) << 3);
#pragma unroll
    for (int v = 0; v < 8; ++v) {
        int row = rbase + v, col = c0 + m;
        size_t off = (size_t)p * HW + (size_t)row * W_IMG + col;
        float xv = out[off];
        float glow = acc[v] * 1.2f;
        float xo = clip01(1.f - (1.f - xv) * (1.f - glow * o_amt));
        out[off] = passthru ? xv : xo;
    }
}

// ---------------------------------------------------------------------------
extern "C" void kernel_launch(void* const* d_in, const int* in_sizes, int n_in,
                              void* d_out, int out_size, void* d_ws, size_t ws_size,
                              hipStream_t stream) {
    const float* x          = (const float*)d_in[0];
    const float* exposure   = (const float*)d_in[1];
    const float* contrast   = (const float*)d_in[2];
    const float* gamma      = (const float*)d_in[3];
    const float* hue_shifts = (const float*)d_in[4];
    const float* sat_mults  = (const float*)d_in[5];
    const float* lum_shifts = (const float*)d_in[6];
    const float* saturation = (const float*)d_in[7];
    const float* vibrance   = (const float*)d_in[8];
    const float* dehaze     = (const float*)d_in[9];
    const float* clarity    = (const float*)d_in[10];
    const float* texture    = (const float*)d_in[11];
    const float* sharpen    = (const float*)d_in[12];
    const float* orton      = (const float*)d_in[13];
    float* out = (float*)d_out;

    // workspace: B0 = X (3ch, later reused as h51 temp), B1 = luma/luma2, B2/B3 = conv temps
    float* B0 = (float*)d_ws;
    float* B1 = B0 + (size_t)3 * B_IMG * HW;
    float* B2 = B1 + (size_t)B_IMG * HW;
    float* B3 = B2 + (size_t)B_IMG * HW;

    size_t npx = (size_t)B_IMG * HW;
    int pw_blocks = (int)((npx + 255) / 256);
    k1_pointwise<<<pw_blocks, 256, 0, stream>>>(x, exposure, contrast, gamma,
                                                hue_shifts, sat_mults, lum_shifts,
                                                saturation, vibrance, dehaze, B0, B1);

    int segs = W_IMG / 256;
    int hb_luma = B_IMG * H_IMG * segs;                      // 12288 blocks
    hconv_dual_kernel<<<hb_luma, 256, 0, stream>>>(B1, B2, B3);

    int tiles_luma = B_IMG * (H_IMG / 16) * (W_IMG / 16);    // 12288 tiles (waves)
    k3_clarity<<<tiles_luma / 8, 256, 0, stream>>>(B2, B3, B1, B0, clarity, texture);

    hconv_kernel<7><<<hb_luma, 256, 0, stream>>>(B1, B2, 1.5f);

    k5_sharpen<<<tiles_luma / 8, 256, 0, stream>>>(B2, B1, B0, out, sharpen);

    int hb_rgb = 3 * B_IMG * H_IMG * segs;                   // 36864 blocks
    hconv_kernel<51><<<hb_rgb, 256, 0, stream>>>(out, B0, 15.f);

    int tiles_rgb = 3 * tiles_luma;                          // 36864 tiles
    k7_orton<<<tiles_rgb / 8, 256, 0, stream>>>(B0, out, orton);
}